// GroupedQueryAttention_18683107737851
// MI455X (gfx1250) — compile-verified
//
#include <hip/hip_runtime.h>
#include <hip/hip_bf16.h>
#include <math.h>

// ---------------------------------------------------------------------------
// GroupedQueryAttention forward for MI455X (gfx1250), fp32, WMMA 16x16x4 F32.
//   B=2, S=2048, E=1024, H=16, KVH=4, D=64, scale = 1/8
// Outputs (concatenated in d_out): attn_output [B,S,E], attn_weights [B,H,S,S]
//
// Pipeline:
//   1) gqa_gemm_nn x3 : Q = x@Wq * scale, K = x@Wk, V = x@Wv   (LDS-staged B)
//   2) gqa_attn_fused : per (b,h,16 q-rows) -> scores into LDS (128 KB panel),
//      softmax in LDS (wave32 shfl reductions), normalized weights written to
//      HBM once, PV GEMM reads weights back from LDS (not HBM).
// ---------------------------------------------------------------------------

#define Bsz   2
#define Sdim  2048
#define Edim  1024
#define Hh    16
#define KVH   4
#define Dd    64
#define NREP  (Hh / KVH)
#define SCALE 0.125f

#define SPAD  (Sdim + 2)          // padded row stride (mod 64 == 2 -> no bank conflicts)
#define OBASE (16 * SPAD)         // float offset of partial-O buffer in smem
#define SMEM_FLOATS (OBASE + 2 * 16 * Dd)
#define SMEM_BYTES  (SMEM_FLOATS * 4)

typedef float v2f __attribute__((ext_vector_type(2)));
typedef float v8f __attribute__((ext_vector_type(8)));

__device__ __forceinline__ v8f wmma_f32(v2f a, v2f b, v8f c) {
  // 8 args: (neg_a, A, neg_b, B, c_mod, C, reuse_a, reuse_b)
  return __builtin_amdgcn_wmma_f32_16x16x4_f32(
      false, a, false, b, (short)0, c, false, false);
}

// ---------------------------------------------------------------------------
// C[M,N] = scale * (A[M,K] @ B[K,N]), row-major. Block = 8 waves; wave w owns
// C tile (blockIdx.y*128 + w*16, blockIdx.x*16). B panel (64 x 16) is staged
// transposed in LDS so fragments are contiguous ds_load_b64.
// ---------------------------------------------------------------------------
__global__ void gqa_gemm_nn(const float* __restrict__ A,
                            const float* __restrict__ B,
                            float* __restrict__ C,
                            int Kdim, int lda, int ldb, int ldc, float scale) {
  const int tid   = threadIdx.x;
  const int lane  = tid & 31;
  const int wave  = tid >> 5;
  const int laneM = lane & 15;           // M (A-row) / N (B-col) index in tile
  const int kh    = (lane >> 4) << 1;    // 0 for lanes 0-15, 2 for lanes 16-31

  const int m0 = blockIdx.y * 128 + wave * 16;
  const int n0 = blockIdx.x * 16;

  __shared__ float Bs[16][66];           // [n][k], 66 stride -> conflict-friendly

  const float* Arow = A + (size_t)(m0 + laneM) * lda;

  v8f c = {};
  for (int k0 = 0; k0 < Kdim; k0 += 64) {
    __syncthreads();
    // cooperative coalesced load + transpose: 64 rows x 16 cols of B
    for (int i = tid; i < 64 * 16; i += 256) {
      const int kr = i >> 4;
      const int n  = i & 15;
      Bs[n][kr] = B[(size_t)(k0 + kr) * ldb + n0 + n];
    }
    __syncthreads();
#pragma unroll
    for (int kk = 0; kk < 64; kk += 4) {
      v2f a   = *(const v2f*)(Arow + k0 + kk + kh);
      v2f bfr = *(const v2f*)(&Bs[laneM][kk + kh]);
      c = wmma_f32(a, bfr, c);
    }
  }

  const int rbase = m0 + ((lane >> 4) << 3);
  float* Cp = C + (size_t)rbase * ldc + n0 + laneM;
#pragma unroll
  for (int g = 0; g < 8; ++g) Cp[(size_t)g * ldc] = c[g] * scale;
}

// ---------------------------------------------------------------------------
// Fused scores + softmax + PV per (b, h, 16 query rows).
//   Phase 1: P[16][2048] = Q_tile @ K^T into LDS (8 waves x 16 k-tiles each)
//   Phase 2: row softmax in LDS (2 rows/wave, shfl_xor reductions), write
//            normalized weights to d_out once.
//   Phase 3: O_tile[16][64] = P @ V ; 8 waves = 4 d-tiles x 2 k-halves,
//            partials combined through LDS.
// ---------------------------------------------------------------------------
__global__ void gqa_attn_fused(const float* __restrict__ Q,
                               const float* __restrict__ Kt,
                               const float* __restrict__ Vt,
                               float* __restrict__ Wout,
                               float* __restrict__ Oout) {
  extern __shared__ float smem[];
  float* Sbuf = smem;            // [16][SPAD]
  float* Obuf = smem + OBASE;    // [2][16][Dd]

  const int tid   = threadIdx.x;
  const int lane  = tid & 31;
  const int wave  = tid >> 5;
  const int laneM = lane & 15;
  const int kh    = (lane >> 4) << 1;

  const int bh  = blockIdx.y;          // b*H + h
  const int b   = bh >> 4;             // H == 16
  const int h   = bh & 15;
  const int kvh = h >> 2;              // NREP == 4
  const int q0  = blockIdx.x * 16;

  // ---- Phase 1: scores into LDS ----
  const float* Qrow = Q + ((size_t)b * Sdim + q0 + laneM) * Edim + h * Dd;
  v2f afr[16];
#pragma unroll
  for (int t = 0; t < 16; ++t) afr[t] = *(const v2f*)(Qrow + 4 * t + kh);

  for (int kt = wave; kt < Sdim / 16; kt += 8) {
    const float* Krow =
        Kt + ((size_t)b * Sdim + kt * 16 + laneM) * (KVH * Dd) + kvh * Dd;
    v8f c = {};
#pragma unroll
    for (int t = 0; t < 16; ++t) {
      v2f bfr = *(const v2f*)(Krow + 4 * t + kh);
      c = wmma_f32(afr[t], bfr, c);
    }
    const int rbase = (lane >> 4) << 3;
    float* Sp = Sbuf + (size_t)rbase * SPAD + kt * 16 + laneM;
#pragma unroll
    for (int g = 0; g < 8; ++g) Sp[(size_t)g * SPAD] = c[g];
  }
  __syncthreads();

  // ---- Phase 2: softmax (2 rows per wave), write weights to HBM once ----
  float* Wbase = Wout + ((size_t)bh * Sdim + q0) * Sdim;
#pragma unroll
  for (int rr = 0; rr < 2; ++rr) {
    const int r = wave * 2 + rr;
    float* row = Sbuf + (size_t)r * SPAD;

    float m = -INFINITY;
    for (int j = lane; j < Sdim; j += 32) m = fmaxf(m, row[j]);
#pragma unroll
    for (int off = 16; off > 0; off >>= 1) m = fmaxf(m, __shfl_xor(m, off, 32));

    float s = 0.f;
    for (int j = lane; j < Sdim; j += 32) {
      float e = __expf(row[j] - m);
      row[j] = e;
      s += e;
    }
#pragma unroll
    for (int off = 16; off > 0; off >>= 1) s += __shfl_xor(s, off, 32);

    const float inv = 1.f / s;
    float* wr = Wbase + (size_t)r * Sdim;
    for (int j = lane; j < Sdim; j += 32) {
      const float v = row[j] * inv;
      row[j] = v;          // normalized weights stay in LDS for Phase 3
      wr[j]  = v;          // single HBM write of attn_weights
    }
  }
  __syncthreads();

  // ---- Phase 3: O = P @ V ; wave -> (d-tile = wave&3, k-half = wave>>2) ----
  const int d0  = (wave & 3) * 16;
  const int k0h = (wave >> 2) * (Sdim / 2);
  const float* Vcol =
      Vt + (size_t)b * Sdim * (KVH * Dd) + kvh * Dd + d0 + laneM;

  v8f c = {};
  for (int kk = k0h; kk < k0h + Sdim / 2; kk += 4) {
    v2f a = *(const v2f*)(Sbuf + (size_t)laneM * SPAD + kk + kh);  // ds_load_b64
    v2f bfr;
    bfr.x = Vcol[(size_t)(kk + kh) * (KVH * Dd)];
    bfr.y = Vcol[(size_t)(kk + kh + 1) * (KVH * Dd)];
    c = wmma_f32(a, bfr, c);
  }
  {
    const int rbase = (lane >> 4) << 3;
    float* Op = Obuf + (size_t)(wave >> 2) * 16 * Dd + (size_t)rbase * Dd + d0 + laneM;
#pragma unroll
    for (int g = 0; g < 8; ++g) Op[(size_t)g * Dd] = c[g];
  }
  __syncthreads();

  // combine the two k-half partials and scatter to [B,S,H,D]
  for (int i = tid; i < 16 * Dd; i += 256) {
    const int r = i >> 6;
    const int d = i & (Dd - 1);
    Oout[((size_t)b * Sdim + q0 + r) * Edim + h * Dd + d] =
        Obuf[i] + Obuf[16 * Dd + i];
  }
}

// ---------------------------------------------------------------------------
extern "C" void kernel_launch(void* const* d_in, const int* in_sizes, int n_in,
                              void* d_out, int out_size, void* d_ws, size_t ws_size,
                              hipStream_t stream) {
  const float* x  = (const float*)d_in[0];  // [B,S,E]
  const float* Wq = (const float*)d_in[1];  // [E,E]
  const float* Wk = (const float*)d_in[2];  // [E,KVH*D]
  const float* Wv = (const float*)d_in[3];  // [E,KVH*D]
  // d_in[4] = Wo is unused by the reference.

  float* Q  = (float*)d_ws;                              // [B*S, 1024]
  float* Kt = Q  + (size_t)Bsz * Sdim * Edim;            // [B*S, 256]
  float* Vt = Kt + (size_t)Bsz * Sdim * (KVH * Dd);      // [B*S, 256]

  float* attn_out = (float*)d_out;                                   // [B,S,E]
  float* weights  = attn_out + (size_t)Bsz * Sdim * Edim;            // [B,H,S,S]

  const int M = Bsz * Sdim;  // 4096

  // Projections: Q (scaled), K, V
  {
    dim3 grid(Edim / 16, M / 128);
    gqa_gemm_nn<<<grid, 256, 0, stream>>>(x, Wq, Q, Edim, Edim, Edim, Edim, SCALE);
  }
  {
    dim3 grid((KVH * Dd) / 16, M / 128);
    gqa_gemm_nn<<<grid, 256, 0, stream>>>(x, Wk, Kt, Edim, Edim, KVH * Dd, KVH * Dd, 1.0f);
    gqa_gemm_nn<<<grid, 256, 0, stream>>>(x, Wv, Vt, Edim, Edim, KVH * Dd, KVH * Dd, 1.0f);
  }

  // Fused scores + softmax + PV
  {
    dim3 grid(Sdim / 16, Bsz * Hh);
    gqa_attn_fused<<<grid, 256, SMEM_BYTES, stream>>>(Q, Kt, Vt, weights, attn_out);
  }
}